// ConcatAttention_29205777613489
// MI455X (gfx1250) — compile-verified
//
#include <hip/hip_runtime.h>
#include <math.h>

// Problem dims (fixed by the reference).
#define BB   4
#define SS   512
#define HH   512
#define AA   128
#define N2   256          // 2*A output columns of the fused [P|Q] GEMM
#define MM   (BB*SS)      // 2048 rows
#define JT   64           // Q rows per LDS tile in the scores kernel

typedef __attribute__((ext_vector_type(2))) float v2f;
typedef __attribute__((ext_vector_type(8))) float v8f;

// ---- hardware tanh (gfx1250 v_tanh_f32 transcendental) ----
#if defined(__has_builtin) && __has_builtin(__builtin_amdgcn_tanhf)
#pragma message("hw_tanh: using __builtin_amdgcn_tanhf")
__device__ __forceinline__ float hw_tanh(float x) { return __builtin_amdgcn_tanhf(x); }
#else
#pragma message("hw_tanh: using inline asm v_tanh_f32")
__device__ __forceinline__ float hw_tanh(float x) {
  float r;
  asm volatile("v_tanh_f32 %0, %1" : "=v"(r) : "v"(x));
  return r;
}
#endif

// ---- async global->LDS copy (CDNA5 GLOBAL_LOAD_ASYNC_TO_LDS_B128) ----
#if defined(__has_builtin) && __has_builtin(__builtin_amdgcn_global_load_async_to_lds_b128)
#define HAVE_ASYNC_LDS 1
#pragma message("staging: using __builtin_amdgcn_global_load_async_to_lds_b128")
typedef __attribute__((__vector_size__(4 * sizeof(int)))) int i128v;
typedef __attribute__((address_space(1))) i128v gi128v;   // global int4
typedef __attribute__((address_space(3))) i128v li128v;   // LDS int4
__device__ __forceinline__ void async_cp16(const float* g, float* l) {
  __builtin_amdgcn_global_load_async_to_lds_b128(
      (gi128v*)(unsigned long long)g, (li128v*)(unsigned long long)l,
      /*imm offset*/0, /*cpol*/0);
}
#if defined(__has_builtin) && __has_builtin(__builtin_amdgcn_s_wait_asynccnt)
__device__ __forceinline__ void wait_async0() { __builtin_amdgcn_s_wait_asynccnt(0); }
#else
__device__ __forceinline__ void wait_async0() {
  asm volatile("s_wait_asynccnt 0x0" ::: "memory");
}
#endif
#else
#define HAVE_ASYNC_LDS 0
#pragma message("staging: async-to-LDS builtin NOT available, sync fallback")
__device__ __forceinline__ void wait_async0() {}
#endif

// Stage one JT x 128 Q tile (32 KB) into LDS: 2048 16B chunks over 256 threads.
__device__ __forceinline__ void stage_tile(float* __restrict__ dst,
                                           const float* __restrict__ src,  // row 0, col 0 of tile (global)
                                           int tid) {
  #pragma unroll
  for (int t = 0; t < 8; ++t) {
    const int idx = tid + t * 256;        // 0..2047
    const int row = idx >> 5;             // 32 chunks per 128-float row
    const int c4  = (idx & 31) << 2;      // float offset of 16B chunk
#if HAVE_ASYNC_LDS
    async_cp16(src + (size_t)row * N2 + c4, dst + row * 128 + c4);
#else
    *(float4*)(dst + row * 128 + c4) = *(const float4*)(src + (size_t)row * N2 + c4);
#endif
  }
}

// -----------------------------------------------------------------------------
// Kernel 1: fused GEMM  PQ[m, n] = sum_h X[m,h] * W(n)[h]  (+ bias for n < A)
//   n <  A : W row = W_w[n,      0..H)   (-> P, bias added)
//   n >= A : W row = W_w[n-A,  H..2H)    (-> Q)
// One wave per 16x16 tile, V_WMMA_F32_16X16X4_F32, two interleaved K-chains.
// -----------------------------------------------------------------------------
__global__ __launch_bounds__(32) void gemm_pq_kernel(const float* __restrict__ X,
                                                     const float* __restrict__ Ww,
                                                     const float* __restrict__ Wb,
                                                     float* __restrict__ PQ) {
  const int tileM = blockIdx.x;            // 0..127
  const int tileN = blockIdx.y;            // 0..15
  const int lane  = threadIdx.x;           // 0..31
  const int half  = lane >> 4;             // 0 or 1
  const int l15   = lane & 15;

  const int mrow  = tileM * 16 + l15;      // A-frag row
  const int nglob = tileN * 16 + l15;      // B-frag column
  const int koff  = half * 2;              // lanes 0-15 -> K0,K1; lanes 16-31 -> K2,K3

  const float* xrow = X + (size_t)mrow * HH;
  const float* wrow = (nglob < AA) ? (Ww + (size_t)nglob * (2 * HH))
                                   : (Ww + (size_t)(nglob - AA) * (2 * HH) + HH);

  v8f c0 = {}, c1 = {};
  for (int k = 0; k < HH; k += 8) {
    v2f a0 = *(const v2f*)(xrow + k + koff);
    v2f b0 = *(const v2f*)(wrow + k + koff);
    c0 = __builtin_amdgcn_wmma_f32_16x16x4_f32(false, a0, false, b0, (short)0, c0, false, false);
    v2f a1 = *(const v2f*)(xrow + k + 4 + koff);
    v2f b1 = *(const v2f*)(wrow + k + 4 + koff);
    c1 = __builtin_amdgcn_wmma_f32_16x16x4_f32(false, a1, false, b1, (short)0, c1, false, false);
  }

  // C/D layout: VGPR r -> M = r (lanes 0-15) or r+8 (lanes 16-31); N = l15.
  const float bias = (nglob < AA) ? Wb[nglob] : 0.0f;
  #pragma unroll
  for (int r = 0; r < 8; ++r) {
    const int row = tileM * 16 + r + half * 8;
    PQ[(size_t)row * N2 + nglob] = c0[r] + c1[r] + bias;
  }
}

// -----------------------------------------------------------------------------
// Kernel 2: scores[i] = sum_j sum_a v[a] * tanh(P[i,a] + Q[j,a])      (i = b*S+s)
// 8 waves / workgroup, one wave per i. Lane L owns channels 4L..4L+3 (the
// a-reduction is order independent), so each j is one ds_load_b128.
// Q tiles double-buffered in LDS via async global->LDS copies.
// -----------------------------------------------------------------------------
__global__ __launch_bounds__(256) void scores_kernel(const float* __restrict__ PQ,
                                                     const float* __restrict__ vw,
                                                     float* __restrict__ scores) {
  __shared__ float qs[2][JT * 128];        // 2 x 32 KB
  const int tid  = threadIdx.x;
  const int wave = tid >> 5;
  const int lane = tid & 31;
  const int i    = blockIdx.x * 8 + wave;  // 8 | 512 -> all waves in WG share batch b
  const int b    = i >> 9;
  const float* qsrc = PQ + (size_t)(b * SS) * N2 + AA;   // Q[b] base (row stride N2)

  const float4 p = *(const float4*)(PQ + (size_t)i * N2 + 4 * lane);
  const float4 v = *(const float4*)(vw + 4 * lane);

  // Prime buffer 0 with the first tile.
  stage_tile(qs[0], qsrc, tid);
  wait_async0();
  __syncthreads();

  float a0 = 0.f, a1 = 0.f, a2 = 0.f, a3 = 0.f;   // 4 independent trans-op chains
  int buf = 0;
  for (int jt = 0; jt < SS; jt += JT) {
    if (jt + JT < SS) stage_tile(qs[buf ^ 1], qsrc + (size_t)(jt + JT) * N2, tid);

    const float* qb = qs[buf];
    #pragma unroll 4
    for (int jj = 0; jj < JT; ++jj) {
      const float4 q = *(const float4*)(qb + jj * 128 + 4 * lane);
      a0 += v.x * hw_tanh(p.x + q.x);
      a1 += v.y * hw_tanh(p.y + q.y);
      a2 += v.z * hw_tanh(p.z + q.z);
      a3 += v.w * hw_tanh(p.w + q.w);
    }

    wait_async0();          // next tile landed (this thread's share)
    __syncthreads();        // everyone's share landed; everyone done reading qs[buf]
    buf ^= 1;
  }

  float acc = (a0 + a1) + (a2 + a3);
  #pragma unroll
  for (int o = 16; o > 0; o >>= 1) acc += __shfl_xor(acc, o, 32);
  if (lane == 0) scores[i] = acc;
}

// -----------------------------------------------------------------------------
// Kernel 3: masked softmax per batch row -> attn  (one workgroup per b)
// -----------------------------------------------------------------------------
__global__ __launch_bounds__(512) void softmax_kernel(const float* __restrict__ scores,
                                                      const int* __restrict__ lengths,
                                                      float* __restrict__ attn) {
  __shared__ float red[512];
  const int b = blockIdx.x;
  const int s = threadIdx.x;
  const int len = lengths[b];
  const float sc = (s < len) ? scores[b * SS + s] : -1000000000.0f;

  red[s] = sc;
  __syncthreads();
  #pragma unroll
  for (int o = 256; o > 0; o >>= 1) {
    if (s < o) red[s] = fmaxf(red[s], red[s + o]);
    __syncthreads();
  }
  const float mx = red[0];
  __syncthreads();

  const float e = __expf(sc - mx);
  red[s] = e;
  __syncthreads();
  #pragma unroll
  for (int o = 256; o > 0; o >>= 1) {
    if (s < o) red[s] += red[s + o];
    __syncthreads();
  }
  attn[b * SS + s] = e / red[0];
}

// -----------------------------------------------------------------------------
// Kernel 4: context[b,h] = sum_s X[b,s,h] * attn[b,s]
// -----------------------------------------------------------------------------
__global__ __launch_bounds__(128) void context_kernel(const float* __restrict__ X,
                                                      const float* __restrict__ attn,
                                                      float* __restrict__ ctx) {
  __shared__ float aS[SS];
  const int b = blockIdx.x;
  const int h = blockIdx.y * 128 + threadIdx.x;
  for (int s = threadIdx.x; s < SS; s += 128) aS[s] = attn[b * SS + s];
  __syncthreads();
  float acc = 0.f;
  for (int s = 0; s < SS; ++s) acc += X[(size_t)(b * SS + s) * HH + h] * aS[s];
  ctx[b * HH + h] = acc;
}

extern "C" void kernel_launch(void* const* d_in, const int* in_sizes, int n_in,
                              void* d_out, int out_size, void* d_ws, size_t ws_size,
                              hipStream_t stream) {
  const float* X       = (const float*)d_in[0];   // (B,S,H) f32
  const int*   lengths = (const int*)d_in[1];     // (B,)    int
  const float* Ww      = (const float*)d_in[2];   // (A,2H)  f32
  const float* Wb      = (const float*)d_in[3];   // (A,)    f32
  const float* vw      = (const float*)d_in[4];   // (A,)    f32

  float* out  = (float*)d_out;
  float* ctx  = out;               // (B,H) = 2048 floats
  float* attn = out + BB * HH;     // (B,S) = 2048 floats

  float* PQ     = (float*)d_ws;    // (M, 2A) = 2 MB
  float* scores = PQ + (size_t)MM * N2;

  gemm_pq_kernel<<<dim3(MM / 16, N2 / 16), 32, 0, stream>>>(X, Ww, Wb, PQ);
  scores_kernel<<<dim3(MM / 8), 256, 0, stream>>>(PQ, vw, scores);
  softmax_kernel<<<dim3(BB), 512, 0, stream>>>(scores, lengths, attn);
  context_kernel<<<dim3(BB, HH / 128), 128, 0, stream>>>(X, attn, ctx);
}